// NEURAL_PYSCF_WF_28776280883249
// MI455X (gfx1250) — compile-verified
//
#include <hip/hip_runtime.h>

// NEURAL_PYSCF_WF: dets of gathered 16x16 minors of per-batch MO matrix,
// ci-weighted sum. f32 end-to-end (matches reference precision).
//
// Phase 1 per batch: mo_sub(32x32) = ao[b](32x128) @ W[0:32,:]^T via
//                    v_wmma_f32_16x16x4_f32 (one 16x16 tile per wave).
// Phase 2 per batch: 128 lanes x 1 config each: gather 16x16 from LDS,
//                    fully-unrolled register LU, det = prod(pivots).
// Phase 3: block reduction -> out[b].

typedef __attribute__((ext_vector_type(2))) float v2f;
typedef __attribute__((ext_vector_type(8))) float v8f;

#define NELEC   32
#define NAO     128
#define NCONF   128
#define KDET    16
#define BPB     8        // batches per block
#define THREADS 128      // 4 waves (wave32)
#define LDW     129      // padded stride for 128-wide rows (bank-conflict-free)
#define LDM     33       // padded stride for 32-wide mo_sub rows

__global__ __launch_bounds__(THREADS)
void NEURAL_PYSCF_WF_28776280883249_kernel(
    const float* __restrict__ ao,     // (8192, 32, 128)
    const float* __restrict__ w,      // (128, 128) -- only rows 0..31 reachable
    const float* __restrict__ ci,     // (1, 128)
    const int*   __restrict__ cfg,    // (128, 16), values in [0,32)
    float*       __restrict__ out)    // (8192,)
{
    __shared__ float w_lds [NELEC * LDW];   // W rows 0..31, padded
    __shared__ float ao_lds[NELEC * LDW];   // ao[b], padded
    __shared__ float mo_lds[NELEC * LDM];   // mo_sub 32x32, padded
    __shared__ int   cfg_lds[NCONF * KDET];
    __shared__ float red[THREADS];

    const int tid  = threadIdx.x;
    const int lane = tid & 31;
    const int wave = tid >> 5;

    // ---- once per block: stage W[0:32,:] (contiguous 4096 f32) and configs ----
    {
        const float4* src = (const float4*)w;
        #pragma unroll
        for (int i = 0; i < 8; ++i) {
            int f4 = tid + i * THREADS;              // 0..1023 float4s
            float4 v = src[f4];
            int row = f4 >> 5, col = (f4 & 31) << 2; // 32 float4 per row
            float* dst = &w_lds[row * LDW + col];
            dst[0] = v.x; dst[1] = v.y; dst[2] = v.z; dst[3] = v.w;
        }
        const int4* csrc = (const int4*)cfg;
        #pragma unroll
        for (int i = 0; i < 4; ++i) {
            int q = tid + i * THREADS;               // 0..511 int4s
            int4 v = csrc[q];
            int* dst = &cfg_lds[q << 2];
            dst[0] = v.x; dst[1] = v.y; dst[2] = v.z; dst[3] = v.w;
        }
    }

    for (int bi = 0; bi < BPB; ++bi) {
        const int b = blockIdx.x * BPB + bi;
        __syncthreads();   // protect ao_lds reuse across iterations

        // ---- stage ao[b] (contiguous 16 KB) into LDS, padded rows ----
        const float4* asrc = (const float4*)(ao + (size_t)b * NELEC * NAO);
        #pragma unroll
        for (int i = 0; i < 8; ++i) {
            int f4 = tid + i * THREADS;
            float4 v = asrc[f4];
            int row = f4 >> 5, col = (f4 & 31) << 2;
            float* dst = &ao_lds[row * LDW + col];
            dst[0] = v.x; dst[1] = v.y; dst[2] = v.z; dst[3] = v.w;
        }
        // prefetch next batch's ao into GL2 while we compute (global_prefetch_b8)
        if (bi + 1 < BPB) {
            const char* nxt = (const char*)(ao + (size_t)(b + 1) * NELEC * NAO);
            __builtin_prefetch(nxt + tid * 128, 0, 1);
        }
        __syncthreads();

        // ---- Phase 1: mo_sub tile (m0,n0) per wave via WMMA f32 16x16x4 ----
        {
            const int m0 = (wave & 1) << 4;          // 0 or 16 (electron rows)
            const int n0 = (wave >> 1) << 4;         // 0 or 16 (MO cols)
            const int r  = lane & 15;
            const int kh = (lane >> 4) << 1;         // lanes 0-15: K{0,1}; 16-31: K{2,3}
            v8f acc = {};
            #pragma unroll
            for (int k0 = 0; k0 < NAO; k0 += 4) {
                const float* ap = &ao_lds[(m0 + r) * LDW + k0 + kh];
                const float* bp = &w_lds [(n0 + r) * LDW + k0 + kh];
                v2f af; af.x = ap[0]; af.y = ap[1];  // A[m][k], A[m][k+1]
                v2f bf; bf.x = bp[0]; bf.y = bp[1];  // B[k][n]=W[n][k], B[k+1][n]
                acc = __builtin_amdgcn_wmma_f32_16x16x4_f32(
                        false, af, false, bf, (short)0, acc, false, false);
            }
            // C layout: VGPR p -> M = m0 + p (+8 for lanes 16-31), N = n0 + (lane&15)
            const int chalf = lane >> 4;
            const int ccol  = n0 + (lane & 15);
            #pragma unroll
            for (int p = 0; p < 8; ++p)
                mo_lds[(m0 + p + (chalf << 3)) * LDM + ccol] = acc[p];
        }
        __syncthreads();

        // ---- Phase 2: one 16x16 determinant per lane (config c = tid) ----
        {
            const int c = tid;
            int idx[KDET];
            #pragma unroll
            for (int i = 0; i < KDET; ++i) idx[i] = cfg_lds[c * KDET + i];

            float a[KDET][KDET];
            #pragma unroll
            for (int i = 0; i < KDET; ++i) {
                const float* rowp = &mo_lds[idx[i] * LDM];
                #pragma unroll
                for (int j = 0; j < KDET; ++j) a[i][j] = rowp[idx[j]];
            }

            float det = 1.0f;
            #pragma unroll
            for (int k = 0; k < KDET; ++k) {
                const float piv = a[k][k];
                det *= piv;
                const float ip = 1.0f / piv;
                #pragma unroll
                for (int i = k + 1; i < KDET; ++i) {
                    const float m = a[i][k] * ip;
                    #pragma unroll
                    for (int j = k + 1; j < KDET; ++j)
                        a[i][j] = fmaf(-m, a[k][j], a[i][j]);
                }
            }
            red[tid] = det * ci[c];
        }
        __syncthreads();

        // ---- Phase 3: reduce 128 partials -> out[b] ----
        if (wave == 0) {
            float s = red[lane] + red[lane + 32] + red[lane + 64] + red[lane + 96];
            #pragma unroll
            for (int off = 16; off > 0; off >>= 1)
                s += __shfl_xor(s, off, 32);
            if (lane == 0) out[b] = s;
        }
    }
}

extern "C" void kernel_launch(void* const* d_in, const int* in_sizes, int n_in,
                              void* d_out, int out_size, void* d_ws, size_t ws_size,
                              hipStream_t stream) {
    (void)in_sizes; (void)n_in; (void)d_ws; (void)ws_size; (void)out_size;
    const float* ao  = (const float*)d_in[0];
    const float* w   = (const float*)d_in[1];
    const float* ci  = (const float*)d_in[2];
    const int*   cfg = (const int*)  d_in[3];
    float*       out = (float*)d_out;

    const int blocks = 8192 / BPB;   // 1024 blocks x 128 threads
    NEURAL_PYSCF_WF_28776280883249_kernel<<<blocks, THREADS, 0, stream>>>(
        ao, w, ci, cfg, out);
}